// GCNEncoder_82162724372661
// MI455X (gfx1250) — compile-verified
//
#include <hip/hip_runtime.h>
#include <hip/hip_bf16.h>

typedef float v2f __attribute__((ext_vector_type(2)));
typedef float v8f __attribute__((ext_vector_type(8)));

#define DIM 128   // IN_DIM == OUT_DIM == 128

// ---------------------------------------------------------------- degree init
__global__ void deg_init_kernel(float* __restrict__ deg, int N) {
  int i = blockIdx.x * blockDim.x + threadIdx.x;
  if (i < N) deg[i] = 1.0f;  // self loop contributes 1 to every node's degree
}

// ------------------------------------------------------------- degree count
__global__ void deg_count_kernel(const long long* __restrict__ dst,
                                 float* __restrict__ deg, int E) {
  int e = blockIdx.x * blockDim.x + threadIdx.x;
  if (e < E) unsafeAtomicAdd(&deg[dst[e]], 1.0f);
}

// -------------------------------------------------------------- deg -> rsqrt
__global__ void dinv_kernel(float* __restrict__ deg, int N) {
  int i = blockIdx.x * blockDim.x + threadIdx.x;
  if (i < N) deg[i] = rsqrtf(deg[i]);  // deg >= 1 always (self loops)
}

// --------------------------------------------------- h = x @ W  (f32 WMMA)
// One wave per 16x16 output tile; K=128 swept in 32 steps of V_WMMA_F32_16X16X4_F32.
// Epilogue: h[row][col] = raw, out[row][col] = b[col] + h*dinv[row]^2 (self-loop fused).
__global__ void gemm_wmma_kernel(const float* __restrict__ x,
                                 const float* __restrict__ W,
                                 const float* __restrict__ bias,
                                 const float* __restrict__ dinv,
                                 float* __restrict__ h,
                                 float* __restrict__ out,
                                 int nTileM) {
  int wave = (blockIdx.x * blockDim.x + threadIdx.x) >> 5;
  int lane = threadIdx.x & 31;
  int tm = wave >> 3;        // 8 column tiles (128/16)
  int tn = wave & 7;
  if (tm >= nTileM) return;  // wave-uniform: EXEC stays all-ones for WMMA
  int m0 = tm << 4;
  int n0 = tn << 4;
  int l16 = lane & 15;             // row of A / col of B within tile
  int kh  = (lane >> 4) << 1;      // K pair selected by lane half

  const float* ap = x + (size_t)(m0 + l16) * DIM + kh;
  const float* bp = W + (size_t)kh * DIM + n0 + l16;

  v8f c = {};
#pragma unroll
  for (int k0 = 0; k0 < DIM; k0 += 4) {
    v2f a = *(const v2f*)(ap + k0);                 // A[M][k0+kh], A[M][k0+kh+1]
    v2f bm;
    bm.x = bp[(size_t)k0 * DIM];                    // B[k0+kh  ][N]
    bm.y = bp[(size_t)(k0 + 1) * DIM];              // B[k0+kh+1][N]
    c = __builtin_amdgcn_wmma_f32_16x16x4_f32(false, a, false, bm,
                                              (short)0, c, false, false);
  }

  int col  = n0 + l16;
  int row0 = m0 + ((lane >> 4) << 3);               // lanes 16-31 hold rows M+8..M+15
  float bc = bias[col];
#pragma unroll
  for (int r = 0; r < 8; ++r) {
    int row = row0 + r;
    float v = c[r];
    h[(size_t)row * DIM + col] = v;
    float di = dinv[row];
    out[(size_t)row * DIM + col] = bc + v * di * di;  // bias + self-loop message
  }
}

// --------------------------------------------------------------- edge scatter
// One wave per edge; lane handles channels {lane, lane+32, lane+64, lane+96}.
__global__ void scatter_kernel(const long long* __restrict__ src,
                               const long long* __restrict__ dst,
                               const float* __restrict__ dinv,
                               const float* __restrict__ h,
                               float* __restrict__ out, int E) {
  long long t = (long long)blockIdx.x * blockDim.x + threadIdx.x;
  long long e = t >> 5;
  int lane = (int)(t & 31);
  if (e >= E) return;
  long long s = src[e];
  long long d = dst[e];
  float norm = dinv[s] * dinv[d];
  const float* hp = h + (size_t)s * DIM;
  float* op = out + (size_t)d * DIM;
#pragma unroll
  for (int j = 0; j < 4; ++j) {
    int cidx = lane + (j << 5);
    unsafeAtomicAdd(&op[cidx], hp[cidx] * norm);
  }
}

// -------------------------------------------------------------------- PReLU
__global__ void prelu_kernel(float* __restrict__ out,
                             const float* __restrict__ alpha, size_t total4) {
  size_t i = (size_t)blockIdx.x * blockDim.x + threadIdx.x;
  if (i >= total4) return;
  float4 v = ((float4*)out)[i];
  int cb = (int)((i * 4) & (DIM - 1));
  v.x = v.x > 0.f ? v.x : alpha[cb + 0] * v.x;
  v.y = v.y > 0.f ? v.y : alpha[cb + 1] * v.y;
  v.z = v.z > 0.f ? v.z : alpha[cb + 2] * v.z;
  v.w = v.w > 0.f ? v.w : alpha[cb + 3] * v.w;
  ((float4*)out)[i] = v;
}

// ------------------------------------------------------------------- launch
extern "C" void kernel_launch(void* const* d_in, const int* in_sizes, int n_in,
                              void* d_out, int out_size, void* d_ws, size_t ws_size,
                              hipStream_t stream) {
  const float*     x     = (const float*)d_in[0];
  const long long* ei    = (const long long*)d_in[1];  // int64 [2, E]
  const float*     W     = (const float*)d_in[2];
  const float*     b     = (const float*)d_in[3];
  const float*     alpha = (const float*)d_in[4];
  float* out = (float*)d_out;

  int N = in_sizes[0] / DIM;
  int E = in_sizes[1] / 2;

  // workspace layout: [h: N*128 f32][deg/dinv: N f32]
  float* h   = (float*)d_ws;
  float* deg = h + (size_t)N * DIM;

  deg_init_kernel<<<(N + 255) / 256, 256, 0, stream>>>(deg, N);
  deg_count_kernel<<<(E + 255) / 256, 256, 0, stream>>>(ei + E, deg, E);
  dinv_kernel<<<(N + 255) / 256, 256, 0, stream>>>(deg, N);

  int nTileM = (N + 15) / 16;                    // N=100000 -> 6250, exact
  long long gthreads = (long long)nTileM * 8 * 32;
  gemm_wmma_kernel<<<(unsigned)((gthreads + 255) / 256), 256, 0, stream>>>(
      x, W, b, deg, h, out, nTileM);

  long long sthreads = (long long)E * 32;
  scatter_kernel<<<(unsigned)((sthreads + 255) / 256), 256, 0, stream>>>(
      ei, ei + E, deg, h, out, E);

  size_t total4 = ((size_t)N * DIM) / 4;
  prelu_kernel<<<(unsigned)((total4 + 255) / 256), 256, 0, stream>>>(out, alpha, total4);
}